// EdgePolicyModel_45028437131551
// MI455X (gfx1250) — compile-verified
//
#include <hip/hip_runtime.h>
#include <hip/hip_bf16.h>

typedef __attribute__((ext_vector_type(2))) float v2f;
typedef __attribute__((ext_vector_type(8))) float v8f;

// ---------------- degree / normalization ----------------

__global__ void k_init_deg(float* __restrict__ deg, int N) {
    int i = blockIdx.x * blockDim.x + threadIdx.x;
    if (i < N) deg[i] = 1.0f;  // self-loop
}

__global__ void k_deg_scatter(const int* __restrict__ dst, float* __restrict__ deg, int E) {
    int e = blockIdx.x * blockDim.x + threadIdx.x;
    if (e < E) unsafeAtomicAdd(&deg[dst[e]], 1.0f);
}

__global__ void k_dinv(float* __restrict__ deg, int N) {
    int i = blockIdx.x * blockDim.x + threadIdx.x;
    if (i < N) deg[i] = rsqrtf(deg[i]);  // deg >= 1 always (self-loops)
}

// ---------------- dense transform: p = (h @ W) * dinv, via V_WMMA_F32_16X16X4_F32 ----------------
// One wave handles 16 nodes: A = 16x4 node features, B = W padded to 4x16, D = 16x16 (cols 0..3 valid).
// A layout (f32, K=4): lanes 0-15 -> (K0,K1), lanes 16-31 -> (K2,K3). B mirrors with N across lanes.
// D layout: VGPR r: lanes 0-15 = row r, col=lane; lanes 16-31 = row r+8, col=lane-16.
// Extraction goes through LDS so each node is written back as ONE b128 store with ONE dinv load.

__global__ void k_transform_wmma(const float* __restrict__ h, const float* __restrict__ W,
                                 const float* __restrict__ dinv, float* __restrict__ p, int N) {
    __shared__ float smem[8][16][4];  // [wave][node-in-tile][feature]
    const int lane = threadIdx.x & 31;
    const int wave = threadIdx.x >> 5;
    const long base = ((long)blockIdx.x * (blockDim.x >> 5) + wave) * 16;

    const int n  = lane & 15;          // row index for A load / col index for D
    const int kb = (lane >> 4) << 1;   // 0 or 2
    const bool fullTile = (base + 16 <= (long)N);

    v2f a;
    {
        long node = base + n;
        if (fullTile || node < (long)N) {
            a = *(const v2f*)(h + node * 4 + kb);  // single b64 load (8B aligned)
        } else {
            a.x = 0.0f; a.y = 0.0f;
        }
    }
    v2f b;
    if (n < 4) {
        b.x = W[kb * 4 + n];
        b.y = W[(kb + 1) * 4 + n];
    } else {
        b.x = 0.0f; b.y = 0.0f;
    }

    v8f c = {};
    c = __builtin_amdgcn_wmma_f32_16x16x4_f32(
        /*neg_a=*/false, a, /*neg_b=*/false, b,
        /*c_mod=*/(short)0, c, /*reuse_a=*/false, /*reuse_b=*/false);

    if (n < 4) {  // cols 0..3 carry the real output; transpose via LDS
        const int rowOff = (lane >> 4) * 8;
#pragma unroll
        for (int r = 0; r < 8; ++r) smem[wave][rowOff + r][n] = c[r];
    }
    __syncthreads();

    if (lane < 16) {
        long ni = base + lane;
        if (fullTile || ni < (long)N) {
            const float di = dinv[ni];
            const float4 v = *(const float4*)&smem[wave][lane][0];  // ds_load_b128
            float4 o;
            o.x = v.x * di; o.y = v.y * di; o.z = v.z * di; o.w = v.w * di;
            *(float4*)(p + ni * 4) = o;  // global_store_b128
        }
    }
}

// ---------------- conv: zero accumulator, edge scatter-add, finish ----------------

__global__ void k_zero4(float4* __restrict__ acc, int N) {
    int i = blockIdx.x * blockDim.x + threadIdx.x;
    if (i < N) acc[i] = make_float4(0.f, 0.f, 0.f, 0.f);
}

__global__ void k_scatter(const int* __restrict__ src, const int* __restrict__ dst,
                          const float* __restrict__ p, float* __restrict__ acc, int E) {
    int e = blockIdx.x * blockDim.x + threadIdx.x;
    if (e >= E) return;
    int s = src[e], d = dst[e];
    const float4 pv = *(const float4*)(p + (long)s * 4);
    float* a = acc + (long)d * 4;
    unsafeAtomicAdd(a + 0, pv.x);
    unsafeAtomicAdd(a + 1, pv.y);
    unsafeAtomicAdd(a + 2, pv.z);
    unsafeAtomicAdd(a + 3, pv.w);
}

// out[i] = relu(dinv[i]*(acc[i] + p[i]) + b)   (p[i] = hW[i]*dinv[i] -> self-loop term)
__global__ void k_finish(const float* __restrict__ acc, const float* __restrict__ p,
                         const float* __restrict__ dinv, const float* __restrict__ bias,
                         float* __restrict__ hout, int N) {
    int i = blockIdx.x * blockDim.x + threadIdx.x;
    if (i >= N) return;
    float di = dinv[i];
    float b0 = bias[0], b1 = bias[1], b2 = bias[2], b3 = bias[3];
    const float4 av = *(const float4*)(acc + (long)i * 4);
    const float4 pv = *(const float4*)(p + (long)i * 4);
    float4 o;
    o.x = fmaxf(fmaf(di, av.x + pv.x, b0), 0.f);
    o.y = fmaxf(fmaf(di, av.y + pv.y, b1), 0.f);
    o.z = fmaxf(fmaf(di, av.z + pv.z, b2), 0.f);
    o.w = fmaxf(fmaf(di, av.w + pv.w, b3), 0.f);
    *(float4*)(hout + (long)i * 4) = o;
}

// ---------------- masked-edge pass + compact softmax ----------------

__global__ void k_init_misc(int* __restrict__ counter) {
    if (threadIdx.x == 0 && blockIdx.x == 0) *counter = 0;
}

__global__ void k_edge_final(const int* __restrict__ src, const int* __restrict__ dst,
                             const int* __restrict__ nodep,
                             const float* __restrict__ h, const float* __restrict__ Wr,
                             const float* __restrict__ br,
                             float* __restrict__ score_out, float* __restrict__ targ_out,
                             int* __restrict__ counter, int* __restrict__ list_e,
                             float* __restrict__ list_logit, int cap, int E) {
    int e = blockIdx.x * blockDim.x + threadIdx.x;
    if (e >= E) return;
    int node = *nodep;
    int s = src[e], d = dst[e];
    bool mask = (s == node) || (d == node);
    int t = (s == node) ? d : s;
    score_out[e] = 0.0f;                         // masked softmax -> exact 0 for unmasked
    targ_out[e]  = mask ? (float)t : -1.0f;      // node ids < 2^24: exact in f32
    if (mask) {
        const float4 hv = *(const float4*)(h + (long)t * 4);
        float logit = fmaf(hv.x, Wr[0], fmaf(hv.y, Wr[1], fmaf(hv.z, Wr[2], hv.w * Wr[3]))) + br[0];
        int idx = atomicAdd(counter, 1);
        if (idx < cap) { list_e[idx] = e; list_logit[idx] = logit; }
    }
}

__global__ void k_softmax(const int* __restrict__ counter, const int* __restrict__ list_e,
                          const float* __restrict__ list_logit, float* __restrict__ score_out,
                          int cap) {
    __shared__ float red[256];
    int n = *counter;
    if (n > cap) n = cap;
    int tid = threadIdx.x;

    float mx = -3.402823466e38f;
    for (int i = tid; i < n; i += 256) mx = fmaxf(mx, list_logit[i]);
    red[tid] = mx;
    __syncthreads();
    for (int s = 128; s > 0; s >>= 1) {
        if (tid < s) red[tid] = fmaxf(red[tid], red[tid + s]);
        __syncthreads();
    }
    mx = red[0];
    __syncthreads();

    float sum = 0.f;
    for (int i = tid; i < n; i += 256) sum += expf(list_logit[i] - mx);
    red[tid] = sum;
    __syncthreads();
    for (int s = 128; s > 0; s >>= 1) {
        if (tid < s) red[tid] += red[tid + s];
        __syncthreads();
    }
    float Z = red[0];
    float inv = (Z > 0.f) ? 1.0f / Z : 0.0f;

    for (int i = tid; i < n; i += 256) score_out[list_e[i]] = expf(list_logit[i] - mx) * inv;
}

// ---------------- launch ----------------

extern "C" void kernel_launch(void* const* d_in, const int* in_sizes, int n_in,
                              void* d_out, int out_size, void* d_ws, size_t ws_size,
                              hipStream_t stream) {
    const float* x     = (const float*)d_in[0];
    const int*   ei    = (const int*)d_in[1];
    const int*   nodep = (const int*)d_in[2];
    const float* W1    = (const float*)d_in[3];
    const float* b1    = (const float*)d_in[4];
    const float* W2    = (const float*)d_in[5];
    const float* b2    = (const float*)d_in[6];
    const float* Wr    = (const float*)d_in[7];
    const float* br    = (const float*)d_in[8];

    const int N = in_sizes[0] / 4;
    const int E = in_sizes[1] / 2;
    const int* src = ei;
    const int* dst = ei + E;

    // workspace layout
    char* ws = (char*)d_ws;
    size_t off = 0;
    float* dinv = (float*)(ws + off); off += (size_t)N * 4;
    float* p    = (float*)(ws + off); off += (size_t)N * 16;
    float* acc  = (float*)(ws + off); off += (size_t)N * 16;
    float* h    = (float*)(ws + off); off += (size_t)N * 16;
    int* counter = (int*)(ws + off);  off += 16;
    long cap = 0;
    if (ws_size > off) cap = (long)((ws_size - off) / 8);
    if (cap > (1L << 22)) cap = 1L << 22;
    int*   list_e     = (int*)(ws + off);   off += (size_t)cap * 4;
    float* list_logit = (float*)(ws + off);

    float* score = (float*)d_out;
    float* targ  = score + E;

    const dim3 blk(256);
    const int gN = (N + 255) / 256;
    const int gE = (E + 255) / 256;
    const int gT = (N + 127) / 128;  // 8 waves/block * 16 nodes/wave

    // normalization
    k_init_deg<<<gN, blk, 0, stream>>>(dinv, N);
    k_deg_scatter<<<gE, blk, 0, stream>>>(dst, dinv, E);
    k_dinv<<<gN, blk, 0, stream>>>(dinv, N);

    // layer 1: x -> h
    k_transform_wmma<<<gT, blk, 0, stream>>>(x, W1, dinv, p, N);
    k_zero4<<<gN, blk, 0, stream>>>((float4*)acc, N);
    k_scatter<<<gE, blk, 0, stream>>>(src, dst, p, acc, E);
    k_finish<<<gN, blk, 0, stream>>>(acc, p, dinv, b1, h, N);

    // layer 2: h -> h
    k_transform_wmma<<<gT, blk, 0, stream>>>(h, W2, dinv, p, N);
    k_zero4<<<gN, blk, 0, stream>>>((float4*)acc, N);
    k_scatter<<<gE, blk, 0, stream>>>(src, dst, p, acc, E);
    k_finish<<<gN, blk, 0, stream>>>(acc, p, dinv, b2, h, N);

    // masked edge scoring + compact softmax
    k_init_misc<<<1, 1, 0, stream>>>(counter);
    k_edge_final<<<gE, blk, 0, stream>>>(src, dst, nodep, h, Wr, br,
                                         score, targ, counter, list_e, list_logit,
                                         (int)cap, E);
    k_softmax<<<1, 256, 0, stream>>>(counter, list_e, list_logit, score, (int)cap);
}